// MessageAttentionPassing_52845277610631
// MI455X (gfx1250) — compile-verified
//
#include <hip/hip_runtime.h>
#include <hip/hip_bf16.h>

typedef __attribute__((ext_vector_type(2))) float v2f;
typedef __attribute__((ext_vector_type(8))) float v8f;

// Problem constants (from reference): B=2, C_IN=64, N=128, T=24, HID=64, C_OUT=64
#define NB   2
#define NCH  64      // C_IN == HID == C_OUT
#define NN   128
#define NT   24
#define X_CSTRIDE   (NN*NT)            // 3072 floats between channels of x
#define X_BSTRIDE   (NCH*NN*NT)        // 196608 floats between batches of x
#define A_ISTRIDE   (NN*NT)            // 3072 floats between i-rows of attention
#define A_BSTRIDE   (NN*NN*NT)         // 393216 floats between batches of attention

__device__ __forceinline__ v8f wmma4(v2f a, v2f b, v8f c) {
  // V_WMMA_F32_16X16X4_F32 : D(16x16 f32) = A(16x4 f32) * B(4x16 f32) + C
  return __builtin_amdgcn_wmma_f32_16x16x4_f32(
      /*neg_a=*/false, a, /*neg_b=*/false, b,
      /*c_mod=*/(short)0, c, /*reuse_a=*/false, /*reuse_b=*/false);
}

__global__ __launch_bounds__(256)
void map_fused_kernel(const float* __restrict__ x,
                      const float* __restrict__ att,
                      const float* __restrict__ W1,
                      const float* __restrict__ b1,
                      const float* __restrict__ W2,
                      const float* __restrict__ b2,
                      float* __restrict__ out)
{
  // ldsA holds h (stage 1/2), then is overwritten with (mv + b2) for stage 3.
  // ldsB holds mh (shared across all 8 waves as einsum B operand).
  __shared__ float ldsA[NN * NCH];   // 32 KB
  __shared__ float ldsB[NN * NCH];   // 32 KB

  const int tid  = threadIdx.x;
  const int lane = tid & 31;
  const int lo   = lane & 15;     // column / M-row selector within 16
  const int hi   = lane >> 4;     // K half selector (f32 WMMA A/B layout)
  const int wave = tid >> 5;      // 0..7 -> 16-row M tile

  const int bt = blockIdx.x;      // 48 blocks: one per (b, t)
  const int b  = bt / NT;
  const int t  = bt % NT;

  // ---------------- Stage 1: h = xp @ W1 + b1  (16x64 tile per wave) --------
  // A rows: n = 16*wave + lo for this (b, t); element c at stride X_CSTRIDE.
  const int n_row = 16 * wave + lo;
  const int xbase = b * X_BSTRIDE + n_row * NT + t;

  v8f hc[4];
  #pragma unroll
  for (int j = 0; j < 4; ++j) {
    const float bias = b1[16 * j + lo];
    #pragma unroll
    for (int r = 0; r < 8; ++r) hc[j][r] = bias;
  }

  for (int kk = 0; kk < 16; ++kk) {
    const int k = 4 * kk + 2 * hi;            // lanes 0-15: K=k,k+1 ; 16-31: K=k+2,k+3
    v2f a;
    a.x = x[xbase + (k    ) * X_CSTRIDE];
    a.y = x[xbase + (k + 1) * X_CSTRIDE];
    #pragma unroll
    for (int j = 0; j < 4; ++j) {
      v2f w;
      w.x = W1[(k    ) * NCH + 16 * j + lo];
      w.y = W1[(k + 1) * NCH + 16 * j + lo];
      hc[j] = wmma4(a, w, hc[j]);
    }
  }

  // D-layout -> row-major LDS (each wave owns rows [16*wave, 16*wave+16))
  #pragma unroll
  for (int j = 0; j < 4; ++j)
    #pragma unroll
    for (int r = 0; r < 8; ++r)
      ldsA[(16 * wave + r + 8 * hi) * NCH + 16 * j + lo] = hc[j][r];

  __syncthreads();

  // ---------------- Stage 2: mv = h@W2[:64] + b2 ; mh = h@W2[64:] -----------
  v8f mvc[4], mhc[4];
  #pragma unroll
  for (int j = 0; j < 4; ++j) {
    const float bias = b2[16 * j + lo];
    #pragma unroll
    for (int r = 0; r < 8; ++r) { mvc[j][r] = bias; mhc[j][r] = 0.0f; }
  }

  for (int kk = 0; kk < 16; ++kk) {
    const int k = 4 * kk + 2 * hi;
    // A fragment from LDS: h[16*wave + lo, k..k+1]  (8B-aligned ds_load_b64)
    v2f a = *(const v2f*)&ldsA[(16 * wave + lo) * NCH + k];
    #pragma unroll
    for (int j = 0; j < 4; ++j) {
      v2f wv, wh;
      wv.x = W2[(k          ) * NCH + 16 * j + lo];
      wv.y = W2[(k + 1      ) * NCH + 16 * j + lo];
      wh.x = W2[(k + NCH    ) * NCH + 16 * j + lo];
      wh.y = W2[(k + NCH + 1) * NCH + 16 * j + lo];
      mvc[j] = wmma4(a, wv, mvc[j]);
      mhc[j] = wmma4(a, wh, mhc[j]);
    }
  }

  __syncthreads();   // all h reads done before ldsA is overwritten with mv+b2

  #pragma unroll
  for (int j = 0; j < 4; ++j)
    #pragma unroll
    for (int r = 0; r < 8; ++r) {
      const int row = 16 * wave + r + 8 * hi;
      ldsA[row * NCH + 16 * j + lo] = mvc[j][r];   // mv + b2
      ldsB[row * NCH + 16 * j + lo] = mhc[j][r];   // mh
    }

  __syncthreads();

  // ---------------- Stage 3: out = asum*(mv+b2) + A_t @ mh ------------------
  // Wave handles i-tile rows [16*wave, 16*wave+16), all 64 output channels.
  const int i0    = 16 * wave;
  const int abase = b * A_BSTRIDE + (i0 + lo) * A_ISTRIDE + t;  // + j*NT

  v8f acc[4];
  #pragma unroll
  for (int j = 0; j < 4; ++j)
    #pragma unroll
    for (int r = 0; r < 8; ++r) acc[j][r] = 0.0f;

  float psum = 0.0f;   // partial row-sum of attention (asum), folded into A loads
  for (int kk = 0; kk < 32; ++kk) {
    const int jr = 4 * kk + 2 * hi;
    v2f a;
    a.x = att[abase + (jr    ) * NT];
    a.y = att[abase + (jr + 1) * NT];
    psum += a.x + a.y;
    #pragma unroll
    for (int n = 0; n < 4; ++n) {
      v2f mb;
      mb.x = ldsB[(jr    ) * NCH + 16 * n + lo];
      mb.y = ldsB[(jr + 1) * NCH + 16 * n + lo];
      acc[n] = wmma4(a, mb, acc[n]);
    }
  }

  // asum[i0+lo]: lane l holds K-half sums for row i0+lo; combine the halves.
  const float asum_lane = psum + __shfl_xor(psum, 16, 32);

  // Combine + transpose-store: out[b, c, i, t]
  #pragma unroll
  for (int r = 0; r < 8; ++r) {
    const int row = r + 8 * hi;                       // C-layout row within tile
    const float as = __shfl(asum_lane, row, 32);      // asum[i0+row]
    const int gi = i0 + row;
    #pragma unroll
    for (int n = 0; n < 4; ++n) {
      const int c = 16 * n + lo;
      const float val = acc[n][r] + as * ldsA[gi * NCH + c];
      out[b * X_BSTRIDE + c * X_CSTRIDE + gi * NT + t] = val;
    }
  }
}

extern "C" void kernel_launch(void* const* d_in, const int* in_sizes, int n_in,
                              void* d_out, int out_size, void* d_ws, size_t ws_size,
                              hipStream_t stream) {
  const float* x   = (const float*)d_in[0];
  const float* att = (const float*)d_in[1];
  const float* W1  = (const float*)d_in[2];
  const float* b1  = (const float*)d_in[3];
  const float* W2  = (const float*)d_in[4];
  const float* b2  = (const float*)d_in[5];
  float* out = (float*)d_out;

  dim3 grid(NB * NT);   // 48 blocks: one per (b, t)
  dim3 block(256);      // 8 wave32 per block, one 16-row M tile each
  hipLaunchKernelGGL(map_fused_kernel, grid, block, 0, stream,
                     x, att, W1, b1, W2, b2, out);
}